// MDCNv4_1_45973329936719
// MI455X (gfx1250) — compile-verified
//
#include <hip/hip_runtime.h>
#include <math.h>

// Problem constants (from reference)
#define CIN   128
#define COUT  256
#define Hh    56
#define Ww    56
#define HW    3136          // Hh*Ww
#define NBAT  4
#define NPOS  12544         // NBAT*HW
#define K2    9
#define KDIM  1152          // CIN*K2
#define PRED_STR 28         // 27 pred channels padded to 28

typedef float v2f __attribute__((ext_vector_type(2)));
typedef float v8f __attribute__((ext_vector_type(8)));

// Workspace layout (float offsets)
#define OFF_XT   0                        // x transposed to NHWC: NPOS*CIN
#define OFF_WT   1605632                  // main weight [kpair][n][2]: KDIM*COUT
#define OFF_OWB  (OFF_WT + 294912)        // offset weight padded N=32: (KDIM/2)*64
#define OFF_PRED (OFF_OWB + 36864)        // pred per position: NPOS*PRED_STR
// total = 2,288,640 floats (~9.2 MB)

// ---- x: (B,C,H,W) -> x_t: (B,H,W,C) ----
__global__ void k_transpose_x(const float* __restrict__ x, float* __restrict__ xt) {
    int idx = blockIdx.x * 256 + threadIdx.x;       // exactly NPOS*CIN threads
    int c   = idx & (CIN - 1);
    int pos = idx >> 7;
    int b   = pos / HW;
    int hw  = pos - b * HW;
    xt[idx] = x[(b * CIN + c) * HW + hw];
}

// ---- weight: (COUT,CIN,3,3) -> w_t[kpair*512 + n*2 + par], k = tap*128 + c ----
__global__ void k_transpose_w(const float* __restrict__ w, float* __restrict__ wt) {
    int idx = blockIdx.x * 256 + threadIdx.x;       // exactly KDIM*COUT threads
    int par = idx & 1;
    int n   = (idx >> 1) & (COUT - 1);
    int kp  = idx >> 9;
    int k   = kp * 2 + par;
    int t   = k >> 7;
    int c   = k & (CIN - 1);
    wt[idx] = w[n * KDIM + c * K2 + t];
}

// ---- offset_w: (27,CIN,3,3) -> owb[kpair*64 + n*2 + par], n padded to 32 ----
__global__ void k_transpose_owb(const float* __restrict__ ow, float* __restrict__ owb) {
    int idx = blockIdx.x * 256 + threadIdx.x;       // exactly (KDIM/2)*64 = 36864
    int par = idx & 1;
    int n   = (idx >> 1) & 31;
    int kp  = idx >> 6;
    int k   = kp * 2 + par;
    int t   = k >> 7;
    int c   = k & (CIN - 1);
    owb[idx] = (n < 27) ? ow[n * KDIM + c * K2 + t] : 0.f;
}

// ---- offset conv as WMMA GEMM: M=64/block, N=32 (27 live), K=1152 ----
// 128 threads = 4 waves; wave w owns M-tile w (16 pos) x full N=32 (two 16-tiles).
__global__ void __launch_bounds__(128)
k_offset_wmma(const float* __restrict__ xt, const float* __restrict__ owb,
              const float* __restrict__ ob, float* __restrict__ pred) {
    __shared__ float lds_A[64 * 132];

    const int tid  = threadIdx.x;
    const int wv   = tid >> 5;
    const int lane = tid & 31;
    const int half = lane >> 4;
    const int lm   = lane & 15;

    const int posbase = blockIdx.x * 64;      // 64 | 3136 -> uniform batch
    const int b0      = posbase / HW;
    const int hwbase  = posbase - b0 * HW;
    const int xbase   = b0 * HW * CIN;

    const float bias0 = ob[lm];                               // n = lm (<27)
    const float bias1 = (16 + lm < 27) ? ob[16 + lm] : 0.f;
    v8f c0, c1;
#pragma unroll
    for (int i = 0; i < 8; ++i) { c0[i] = bias0; c1[i] = bias1; }

    for (int t = 0; t < K2; ++t) {
        const int ty = t / 3, tx = t % 3;
        // ---- stage im2col A chunk: 64 positions x 128 channels ----
#pragma unroll
        for (int pass = 0; pass < 4; ++pass) {
            int p  = pass * 16 + (tid >> 3);
            int sl = tid & 7;
            int hw = hwbase + p;
            int h  = hw / Ww;
            int w  = hw - h * Ww;
            int y  = h - 1 + ty;
            int x  = w - 1 + tx;
            float s = ((unsigned)y < (unsigned)Hh && (unsigned)x < (unsigned)Ww) ? 1.f : 0.f;
            int yc = min(max(y, 0), Hh - 1), xc = min(max(x, 0), Ww - 1);
            const float* src = xt + xbase + (yc * Ww + xc) * CIN;
#pragma unroll
            for (int j = 0; j < 4; ++j) {
                int c = sl * 4 + j * 32;
                float4 v = *(const float4*)(src + c);
                float4 r; r.x = s * v.x; r.y = s * v.y; r.z = s * v.z; r.w = s * v.w;
                *(float4*)&lds_A[p * 132 + c] = r;
            }
        }
        __syncthreads();
        // ---- 32 k-steps of WMMA over this tap ----
        const float* arow  = &lds_A[(wv * 16 + lm) * 132 + 2 * half];
        const float* bbase = owb + (size_t)(t * 64 + half) * 64;
#pragma unroll 4
        for (int kk = 0; kk < 32; ++kk) {
            v2f a  = *(const v2f*)(arow + kk * 4);
            v2f b0v = *(const v2f*)(bbase + kk * 128 + lm * 2);
            v2f b1v = *(const v2f*)(bbase + kk * 128 + 32 + lm * 2);
            c0 = __builtin_amdgcn_wmma_f32_16x16x4_f32(false, a, false, b0v, (short)0, c0, false, false);
            c1 = __builtin_amdgcn_wmma_f32_16x16x4_f32(false, a, false, b1v, (short)0, c1, false, false);
        }
        __syncthreads();
    }
    // ---- store pred (stride 28), sigmoid on channels 18..26 ----
#pragma unroll
    for (int r = 0; r < 8; ++r) {
        int m   = r + 8 * half;
        int pos = posbase + wv * 16 + m;
        pred[pos * PRED_STR + lm] = c0[r];                    // n = lm in 0..15 (offsets)
        int n1 = 16 + lm;
        if (n1 < 27) {
            float v = c1[r];
            pred[pos * PRED_STR + n1] = (n1 < 18) ? v : 1.f / (1.f + __expf(-v));
        }
    }
}

// ---- main deformable conv WMMA GEMM: M=32/block, each wave a 32x32 C block ----
// 256 threads = 8 waves; ping-pong LDS A buffers so tap t+1 sampling overlaps tap t WMMAs.
__global__ void __launch_bounds__(256)
k_deform(const float* __restrict__ xt, const float* __restrict__ wt,
         const float* __restrict__ pred, float* __restrict__ out) {
    __shared__ float lds_A[2][32 * 132];

    const int tid  = threadIdx.x;
    const int wv   = tid >> 5;
    const int lane = tid & 31;
    const int half = lane >> 4;
    const int lm   = lane & 15;

    const int posbase = blockIdx.x * 32;      // 32 | 3136 -> uniform batch
    const int b0      = posbase / HW;
    const int hwbase  = posbase - b0 * HW;
    const int xbase   = b0 * HW * CIN;

    // sampler decomposition: 32 positions x 8 channel-lanes (16 ch each)
    const int sp   = tid >> 3;
    const int sl   = tid & 7;
    const int hw_s = hwbase + sp;
    const int h_s  = hw_s / Ww;
    const int w_s  = hw_s - h_s * Ww;
    const float* pr = pred + (posbase + sp) * PRED_STR;

    auto sample = [&](int t, float* buf) {
        float dy = pr[2 * t], dx = pr[2 * t + 1], m = pr[18 + t];
        float py = (float)(h_s - 1 + t / 3) + dy;
        float px = (float)(w_s - 1 + t % 3) + dx;
        float y0f = floorf(py), x0f = floorf(px);
        float ly = py - y0f, lx = px - x0f;
        int y0 = (int)y0f, x0 = (int)x0f;
        int y1 = y0 + 1, x1 = x0 + 1;
        float vy0 = ((unsigned)y0 < (unsigned)Hh) ? 1.f : 0.f;
        float vy1 = ((unsigned)y1 < (unsigned)Hh) ? 1.f : 0.f;
        float vx0 = ((unsigned)x0 < (unsigned)Ww) ? 1.f : 0.f;
        float vx1 = ((unsigned)x1 < (unsigned)Ww) ? 1.f : 0.f;
        float w00 = (1.f - ly) * (1.f - lx) * m * vy0 * vx0;
        float w01 = (1.f - ly) * lx         * m * vy0 * vx1;
        float w10 = ly         * (1.f - lx) * m * vy1 * vx0;
        float w11 = ly         * lx         * m * vy1 * vx1;
        int y0c = min(max(y0, 0), Hh - 1), y1c = min(max(y1, 0), Hh - 1);
        int x0c = min(max(x0, 0), Ww - 1), x1c = min(max(x1, 0), Ww - 1);
        const float* p00 = xt + xbase + (y0c * Ww + x0c) * CIN;
        const float* p01 = xt + xbase + (y0c * Ww + x1c) * CIN;
        const float* p10 = xt + xbase + (y1c * Ww + x0c) * CIN;
        const float* p11 = xt + xbase + (y1c * Ww + x1c) * CIN;
#pragma unroll
        for (int j = 0; j < 4; ++j) {
            int c = sl * 4 + j * 32;
            float4 a00 = *(const float4*)(p00 + c);
            float4 a01 = *(const float4*)(p01 + c);
            float4 a10 = *(const float4*)(p10 + c);
            float4 a11 = *(const float4*)(p11 + c);
            float4 r;
            r.x = w00 * a00.x + w01 * a01.x + w10 * a10.x + w11 * a11.x;
            r.y = w00 * a00.y + w01 * a01.y + w10 * a10.y + w11 * a11.y;
            r.z = w00 * a00.z + w01 * a01.z + w10 * a10.z + w11 * a11.z;
            r.w = w00 * a00.w + w01 * a01.w + w10 * a10.w + w11 * a11.w;
            *(float4*)&buf[sp * 132 + c] = r;
        }
    };

    v8f c00 = {}, c01 = {}, c10 = {}, c11 = {};
    const int n0 = wv * 32 + lm;
    const int n1 = n0 + 16;

    sample(0, &lds_A[0][0]);
    __syncthreads();

    for (int t = 0; t < K2; ++t) {
        if (t < 8) sample(t + 1, &lds_A[(t + 1) & 1][0]);   // overlaps WMMAs below
        const float* base  = &lds_A[t & 1][0];
        const float* a0row = base + lm * 132 + 2 * half;          // M tile 0 (pos 0..15)
        const float* a1row = base + (16 + lm) * 132 + 2 * half;   // M tile 1 (pos 16..31)
        const float* bb    = wt + (size_t)(t * 64 + half) * 512;
#pragma unroll 4
        for (int kk = 0; kk < 32; ++kk) {
            v2f a0  = *(const v2f*)(a0row + kk * 4);
            v2f a1  = *(const v2f*)(a1row + kk * 4);
            v2f b0v = *(const v2f*)(bb + (size_t)kk * 1024 + n0 * 2);
            v2f b1v = *(const v2f*)(bb + (size_t)kk * 1024 + n1 * 2);
            c00 = __builtin_amdgcn_wmma_f32_16x16x4_f32(false, a0, false, b0v, (short)0, c00, false, false);
            c01 = __builtin_amdgcn_wmma_f32_16x16x4_f32(false, a0, false, b1v, (short)0, c01, false, false);
            c10 = __builtin_amdgcn_wmma_f32_16x16x4_f32(false, a1, false, b0v, (short)0, c10, false, false);
            c11 = __builtin_amdgcn_wmma_f32_16x16x4_f32(false, a1, false, b1v, (short)0, c11, false, false);
        }
        __syncthreads();
    }

    // ---- store: out (B, COUT, H, W) ----
#pragma unroll
    for (int r = 0; r < 8; ++r) {
        int m   = r + 8 * half;
        int hw0 = hwbase + m;
        int hw1 = hwbase + 16 + m;
        out[(size_t)(b0 * COUT + n0) * HW + hw0] = c00[r];
        out[(size_t)(b0 * COUT + n1) * HW + hw0] = c01[r];
        out[(size_t)(b0 * COUT + n0) * HW + hw1] = c10[r];
        out[(size_t)(b0 * COUT + n1) * HW + hw1] = c11[r];
    }
}

extern "C" void kernel_launch(void* const* d_in, const int* in_sizes, int n_in,
                              void* d_out, int out_size, void* d_ws, size_t ws_size,
                              hipStream_t stream) {
    (void)in_sizes; (void)n_in; (void)out_size; (void)ws_size;
    const float* x   = (const float*)d_in[0];   // (4,128,56,56)
    const float* ow  = (const float*)d_in[1];   // (27,128,3,3)
    const float* ob  = (const float*)d_in[2];   // (27,)
    const float* wgt = (const float*)d_in[3];   // (256,128,3,3)
    float* ws   = (float*)d_ws;
    float* xt   = ws + OFF_XT;
    float* wt   = ws + OFF_WT;
    float* owb  = ws + OFF_OWB;
    float* pred = ws + OFF_PRED;
    float* out  = (float*)d_out;

    k_transpose_x  <<<6272, 256, 0, stream>>>(x, xt);      // NPOS*CIN / 256
    k_transpose_w  <<<1152, 256, 0, stream>>>(wgt, wt);    // KDIM*COUT / 256
    k_transpose_owb<<<144,  256, 0, stream>>>(ow, owb);    // (KDIM/2)*64 / 256
    k_offset_wmma  <<<196,  128, 0, stream>>>(xt, owb, ob, pred);  // NPOS/64
    k_deform       <<<392,  256, 0, stream>>>(xt, wt, pred, out);  // NPOS/32
}